// matchNetLoss_20529943674946
// MI455X (gfx1250) — compile-verified
//
#include <hip/hip_runtime.h>
#include <hip/hip_bf16.h>
#include <math.h>

typedef __attribute__((ext_vector_type(2))) float v2f;
typedef __attribute__((ext_vector_type(8))) float v8f;

#define BS 16
#define G  512
#define S  20
#define Q  256
#define P  (G * S)            // 10240
#define THRESH 0.01f
#define BIGF 1e10f
#define WAVES_PER_BLOCK 8
#define PTILES (P / 16)       // 640 row tiles per batch
#define BLOCKS_X (PTILES / WAVES_PER_BLOCK)  // 80
#define QT (Q / 16)           // 16 column tiles

// ---------------------------------------------------------------- init ----
__global__ void mnl_init_ws(float* bceSum, float* cd1sum, int* cnt, int* colMinBits) {
    int i = blockIdx.x * blockDim.x + threadIdx.x;
    if (i == 0) *bceSum = 0.0f;
    if (i < BS) { cd1sum[i] = 0.0f; cnt[i] = 0; }
    if (i < BS * Q) colMinBits[i] = __float_as_int(BIGF);
}

// ------------------------------------------------------- BCE + mask cnt ---
__global__ __launch_bounds__(256)
void mnl_bce_cnt(const float* __restrict__ probs, const float* __restrict__ target,
                 float* bceSum, int* cnt) {
    __shared__ float red[256];
    int i = blockIdx.x * 256 + threadIdx.x;           // i < BS*G == 8192
    float p = probs[i];
    float t = target[i];
    float lp  = fmaxf(logf(p),    -100.0f);
    float l1p = fmaxf(log1pf(-p), -100.0f);
    float loss = -(t * lp + (1.0f - t) * l1p);
    int b = i / G;
    if (p > THRESH) atomicAdd(&cnt[b], S);            // each cell contributes S points
    red[threadIdx.x] = loss;
    __syncthreads();
    for (int s = 128; s > 0; s >>= 1) {
        if (threadIdx.x < s) red[threadIdx.x] += red[threadIdx.x + s];
        __syncthreads();
    }
    if (threadIdx.x == 0) atomicAdd(bceSum, red[0]);
}

// ------------------------------------------------- Chamfer via f32 WMMA ---
// d2(p,q) = a2 + b2 - 2 a.b as a single V_WMMA_F32_16X16X4_F32 per tile:
//   A row  = [ax, ay, az, 1]
//   B col  = [-2bx, -2by, -2bz, b2]   (staged in LDS, preloaded to VGPRs)
//   C bias = a2 broadcast along the row  -> D = d2 tile directly.
__global__ __launch_bounds__(256)
void mnl_chamfer(const float* __restrict__ diff_pred,
                 const float* __restrict__ probs,
                 const float* __restrict__ diff_gt,
                 float* cd1sum, int* gColMinBits) {
    __shared__ float sB[Q * 4];       // per column: {-2bx, -2by, -2bz, b2}
    __shared__ int   sColMin[Q];
    const int b   = blockIdx.y;
    const int tid = threadIdx.x;

    // ---- stage B operand & init column minima (256 threads, 256 columns)
    {
        const float* bp = diff_gt + ((size_t)b * Q + tid) * 3;
        const float bx = bp[0], by = bp[1], bz = bp[2];
        sB[tid * 4 + 0] = -2.0f * bx;
        sB[tid * 4 + 1] = -2.0f * by;
        sB[tid * 4 + 2] = -2.0f * bz;
        sB[tid * 4 + 3] = bx * bx + by * by + bz * bz;
        sColMin[tid] = __float_as_int(BIGF);
    }
    __syncthreads();

    const int lane    = tid & 31;
    const int wave    = tid >> 5;
    const int r       = lane & 15;
    const int halfsel = lane >> 4;            // 0: K=0,1 ; 1: K=2,3
    const int p0      = (blockIdx.x * WAVES_PER_BLOCK + wave) * 16;
    const int p       = p0 + r;

    // ---- A tile (ISA layout: lanes 0-15 hold K=0,1 ; lanes 16-31 hold K=2,3)
    const float* ap = diff_pred + ((size_t)b * P + p) * 3;
    const float ax = ap[0], ay = ap[1], az = ap[2];
    const float a2 = ax * ax + ay * ay + az * az;
    const float maskf = (probs[b * G + (p / S)] > THRESH) ? 1.0f : 0.0f;

    v2f A;
    A.x = halfsel ? az   : ax;                // branchless -> v_cndmask
    A.y = halfsel ? 1.0f : ay;

    // ---- preload all 16 B K-pairs to registers (in-order DS returns ->
    //      partial s_wait_dscnt thresholds; WMMAs then issue back-to-back)
    v2f Bt[QT];
#pragma unroll
    for (int qt = 0; qt < QT; ++qt) {
        const int q = qt * 16 + r;
        Bt[qt] = *reinterpret_cast<const v2f*>(&sB[(q << 2) + (halfsel << 1)]);
    }

    // C/D layout: VGPR v, lane half h -> row M = v + 8*h. Gather a2 & mask per M.
    const int off = halfsel << 3;             // 0 or 8
    v8f   Cbias;
    float mv[8];
#pragma unroll
    for (int v = 0; v < 8; ++v) {
        Cbias[v] = __shfl(a2,    v + off, 32);
        mv[v]    = __shfl(maskf, v + off, 32);
    }

    float rowmin[8];
#pragma unroll
    for (int v = 0; v < 8; ++v) rowmin[v] = BIGF;

#pragma unroll
    for (int qt = 0; qt < QT; ++qt) {
        // 8 args: (neg_a, A, neg_b, B, c_mod, C, reuse_a, reuse_b)
        v8f d = __builtin_amdgcn_wmma_f32_16x16x4_f32(
            false, A, false, Bt[qt], (short)0, Cbias, false, false);

        float cc = BIGF;
#pragma unroll
        for (int v = 0; v < 8; ++v) {
            const float d2v = fmaxf(d[v], 0.0f);                 // clip(d2, 0)
            rowmin[v] = fminf(rowmin[v], d2v);                   // min over Q per row
            cc        = fminf(cc, (mv[v] > 0.5f) ? d2v : BIGF);  // masked col min
        }
        // all 32 lanes commit (lane and lane^16 share q -> combined min is exact)
        atomicMin(&sColMin[qt * 16 + r], __float_as_int(cc));    // d2>=0 -> int order ok
    }

    // ---- reduce per-row minima across each 16-lane half, then cd1 partial sum
    float s = 0.0f;
#pragma unroll
    for (int v = 0; v < 8; ++v) {
        float m = rowmin[v];
        m = fminf(m, __shfl_xor(m, 1, 32));
        m = fminf(m, __shfl_xor(m, 2, 32));
        m = fminf(m, __shfl_xor(m, 4, 32));
        m = fminf(m, __shfl_xor(m, 8, 32));
        s += (mv[v] > 0.5f) ? sqrtf(m) : 0.0f;                   // rows v+off, masked
    }
    const float s_hi = __shfl(s, 16, 32);                        // rows 8..15 partial
    if (lane == 0) atomicAdd(&cd1sum[b], s + s_hi);

    __syncthreads();
    atomicMin(&gColMinBits[b * Q + tid], sColMin[tid]);
}

// ------------------------------------------------------------- finalize ---
__global__ void mnl_finalize(const float* bceSum, const float* cd1sum,
                             const int* cnt, const int* colMinBits, float* out) {
    __shared__ float cds[BS];
    const int b = threadIdx.x;
    if (b < BS) {
        float c2 = 0.0f;
        for (int q = 0; q < Q; ++q)
            c2 += sqrtf(fmaxf(__int_as_float(colMinBits[b * Q + q]), 0.0f));
        c2 /= (float)Q;
        const float c1 = cd1sum[b] / fmaxf((float)cnt[b], 1.0f);
        cds[b] = (cnt[b] > 0) ? fmaxf(c1, c2) : 0.0f;
    }
    __syncthreads();
    if (threadIdx.x == 0) {
        float cs = 0.0f;
        for (int i = 0; i < BS; ++i) cs += cds[i];
        const float pred_loss = *bceSum / (float)(BS * G);
        out[0] = pred_loss + cs / (float)BS;
    }
}

// ---------------------------------------------------------------- launch --
extern "C" void kernel_launch(void* const* d_in, const int* in_sizes, int n_in,
                              void* d_out, int out_size, void* d_ws, size_t ws_size,
                              hipStream_t stream) {
    const float* diff_pred   = (const float*)d_in[0];  // [BS,G,S,3]
    const float* probs_pred  = (const float*)d_in[1];  // [BS,G]
    const float* diff_gt     = (const float*)d_in[2];  // [BS,Q,3]
    const float* prob_target = (const float*)d_in[3];  // [BS,G]
    float* out = (float*)d_out;

    float* ws         = (float*)d_ws;
    float* bceSum     = ws;                       // 1 float
    float* cd1sum     = ws + 1;                   // BS floats
    int*   cnt        = (int*)(ws + 1 + BS);      // BS ints
    int*   colMinBits = (int*)(ws + 1 + 2 * BS);  // BS*Q ints

    mnl_init_ws<<<dim3((BS * Q + 255) / 256), dim3(256), 0, stream>>>(
        bceSum, cd1sum, cnt, colMinBits);
    mnl_bce_cnt<<<dim3((BS * G) / 256), dim3(256), 0, stream>>>(
        probs_pred, prob_target, bceSum, cnt);
    mnl_chamfer<<<dim3(BLOCKS_X, BS), dim3(256), 0, stream>>>(
        diff_pred, probs_pred, diff_gt, cd1sum, colMinBits);
    mnl_finalize<<<dim3(1), dim3(32), 0, stream>>>(
        bceSum, cd1sum, cnt, colMinBits, out);
}